// GNNRankTask2_35381940584593
// MI455X (gfx1250) — compile-verified
//
#include <hip/hip_runtime.h>

typedef __attribute__((ext_vector_type(2))) float v2f;
typedef __attribute__((ext_vector_type(8))) float v8f;

#define D 128
#define NPG 512        // nodes per graph
#define ROWP 288       // dwords per K-pair row in LDS; 288 mod 64 == 32 -> halves hit disjoint banks

// ---------------- degree counting ----------------
__global__ void deg_kernel(const int* __restrict__ src, const int* __restrict__ dst,
                           float* __restrict__ dout, float* __restrict__ din, int E) {
  int e = blockIdx.x * blockDim.x + threadIdx.x;
  if (e < E) {
    atomicAdd(&dout[src[e]], 1.0f);
    atomicAdd(&din[dst[e]], 1.0f);
  }
}

__global__ void isqrt_kernel(float* __restrict__ d, int n) {
  int i = blockIdx.x * blockDim.x + threadIdx.x;
  if (i < n) d[i] = rsqrtf(fmaxf(d[i], 1.0f));
}

// ---------------- layer-1 scatter: emb gather fused ----------------
__global__ void scatter_emb_kernel(const int* __restrict__ src, const int* __restrict__ dst,
                                   const int* __restrict__ nf, const float* __restrict__ emb,
                                   const float* __restrict__ oisq, float* __restrict__ agg) {
  int e = blockIdx.x;      // one edge per 128-thread block
  int f = threadIdx.x;     // feature
  int s = src[e];
  int d = dst[e];
  float v = emb[(size_t)nf[s] * D + f] * oisq[s];
  atomicAdd(&agg[(size_t)d * D + f], v);
}

// ---------------- layer-2 scatter: h already pre-scaled by out_isqrt ----------------
__global__ void scatter_h_kernel(const int* __restrict__ src, const int* __restrict__ dst,
                                 const float* __restrict__ h, float* __restrict__ agg) {
  int e = blockIdx.x;
  int f = threadIdx.x;
  atomicAdd(&agg[(size_t)dst[e] * D + f], h[(size_t)src[e] * D + f]);
}

// ---------------- WMMA GEMM + epilogue ----------------
// out_pre = agg @ W              (f32 WMMA, 16x16x4)
// v       = relu(in_isqrt[row] * out_pre + b[col])
// mode 0: hout[row,col] = v * out_isqrt[row]           (feeds layer-2 scatter)
// mode 1: pool[g,col]  += v / NPG                       (fused mean-pool output)
//
// W is staged in LDS K-pair interleaved: WL[p*ROWP + 2*c + s] = W[2p+s][c]
// so each lane's B operand (rows k+2*half, k+2*half+1 at one column) is a single
// contiguous ds_load_b64 -> lands in an adjacent even VGPR pair, no packing moves.
__global__ __launch_bounds__(256)
void gcn_mm_kernel(const float* __restrict__ agg, const float* __restrict__ Wg,
                   const float* __restrict__ bias, const float* __restrict__ iisq,
                   const float* __restrict__ oisq, float* __restrict__ hout,
                   float* __restrict__ pool, int mode) {
  __shared__ float Wl[(D / 2) * ROWP];
  for (int i = threadIdx.x; i < D * D; i += 256) {
    int k = i >> 7;          // K row
    int c = i & 127;         // column
    Wl[(k >> 1) * ROWP + 2 * c + (k & 1)] = Wg[i];
  }
  __syncthreads();

  const int wave = threadIdx.x >> 5;   // 8 waves, each owns a 16-row tile
  const int lane = threadIdx.x & 31;
  const int half = lane >> 4;          // K-half for A/B f32 layout
  const int l    = lane & 15;
  const int rowbase = blockIdx.x * 128 + wave * 16;

  v8f c[8];
  v8f zero = {0.f, 0.f, 0.f, 0.f, 0.f, 0.f, 0.f, 0.f};
#pragma unroll
  for (int nt = 0; nt < 8; ++nt) c[nt] = zero;

  // A layout (16x4 f32): lanes 0-15 hold row l with K = k,k+1; lanes 16-31 K = k+2,k+3
  const float* arow = agg + (size_t)(rowbase + l) * D + 2 * half;

#pragma unroll 4
  for (int kk = 0; kk < 32; ++kk) {
    const v2f av = *(const v2f*)(arow + kk * 4);          // global b64
    // B layout (4x16 f32): lanes 0-15 rows k,k+1; lanes 16-31 rows k+2,k+3
    const float* bp = &Wl[(kk * 2 + half) * ROWP + 2 * l];
#pragma unroll
    for (int nt = 0; nt < 8; ++nt) {
      const v2f bv = *(const v2f*)(bp + nt * 32);          // ds b64, conflict-free
      c[nt] = __builtin_amdgcn_wmma_f32_16x16x4_f32(
          false, av, false, bv, (short)0, c[nt], false, false);
    }
  }

  // epilogue: C/D layout — VGPR r, lanes 0-15: (M=r, N=lane); lanes 16-31: (M=r+8, N=lane-16)
  float srow[8], orow[8];
#pragma unroll
  for (int r = 0; r < 8; ++r) {
    int row = rowbase + r + 8 * half;
    srow[r] = iisq[row];
    orow[r] = (mode == 0) ? oisq[row] : 1.0f;
  }
  const int g = rowbase / NPG;   // 128-row block never straddles a graph (512 % 128 == 0)

#pragma unroll
  for (int nt = 0; nt < 8; ++nt) {
    const int col = nt * 16 + l;
    const float bc = bias[col];
    float colsum = 0.f;
#pragma unroll
    for (int r = 0; r < 8; ++r) {
      float v = fmaxf(srow[r] * c[nt][r] + bc, 0.0f);
      if (mode == 0) {
        int row = rowbase + r + 8 * half;
        hout[(size_t)row * D + col] = v * orow[r];
      } else {
        colsum += v;
      }
    }
    if (mode == 1) {
      colsum += __shfl_xor(colsum, 16, 32);   // combine the two M-halves (wave32)
      if (half == 0) atomicAdd(&pool[g * D + col], colsum * (1.0f / NPG));
    }
  }
}

extern "C" void kernel_launch(void* const* d_in, const int* in_sizes, int n_in,
                              void* d_out, int out_size, void* d_ws, size_t ws_size,
                              hipStream_t stream) {
  const int*   node_feat = (const int*)d_in[0];
  const int*   src       = (const int*)d_in[1];
  const int*   dst       = (const int*)d_in[2];
  const float* emb       = (const float*)d_in[3];
  const float* W1        = (const float*)d_in[4];
  const float* b1        = (const float*)d_in[5];
  const float* W2        = (const float*)d_in[6];
  const float* b2        = (const float*)d_in[7];

  const int Ntot = in_sizes[0];   // 131072
  const int E    = in_sizes[1];   // 2097152

  char* ws = (char*)d_ws;
  const size_t aggBytes = (size_t)Ntot * D * sizeof(float);   // 64 MB
  float* agg  = (float*)(ws);
  float* h1   = (float*)(ws + aggBytes);
  float* oisq = (float*)(ws + 2 * aggBytes);
  float* iisq = oisq + Ntot;
  float* out  = (float*)d_out;

  // zero accumulators (graph-capturable async memsets)
  hipMemsetAsync(agg, 0, aggBytes, stream);
  hipMemsetAsync(oisq, 0, (size_t)2 * Ntot * sizeof(float), stream);
  hipMemsetAsync(out, 0, (size_t)out_size * sizeof(float), stream);

  // degrees -> rsqrt(max(deg,1))  (oisq/iisq are contiguous: one transform launch)
  deg_kernel<<<(E + 255) / 256, 256, 0, stream>>>(src, dst, oisq, iisq, E);
  isqrt_kernel<<<(2 * Ntot + 255) / 256, 256, 0, stream>>>(oisq, 2 * Ntot);

  // layer 1
  scatter_emb_kernel<<<E, D, 0, stream>>>(src, dst, node_feat, emb, oisq, agg);
  gcn_mm_kernel<<<Ntot / 128, 256, 0, stream>>>(agg, W1, b1, iisq, oisq, h1, out, 0);

  // layer 2 (reuse agg)
  hipMemsetAsync(agg, 0, aggBytes, stream);
  scatter_h_kernel<<<E, D, 0, stream>>>(src, dst, h1, agg);
  gcn_mm_kernel<<<Ntot / 128, 256, 0, stream>>>(agg, W2, b2, iisq, oisq, h1, out, 1);
}